// Net_43782896615457
// MI455X (gfx1250) — compile-verified
//
#include <hip/hip_runtime.h>

typedef float v2f __attribute__((ext_vector_type(2)));
typedef float v8f __attribute__((ext_vector_type(8)));

#define HIDDEN 128

// ---------------------------------------------------------------------------
// Kernel A: tiny MLP (2x53 -> relu(128) -> 3) then @ local_frame^T.
// Writes translation_vectors (2x3 = 6 floats) into d_ws.
// ---------------------------------------------------------------------------
__global__ void mlp_kernel(const float* __restrict__ latent,
                           const float* __restrict__ features,
                           const float* __restrict__ W1,
                           const float* __restrict__ b1,
                           const float* __restrict__ W2,
                           const float* __restrict__ b2,
                           const float* __restrict__ local_frame,
                           float* __restrict__ tv_out) {
    __shared__ float h_s[2 * HIDDEN];
    __shared__ float s_s[6];
    const int t = threadIdx.x;                       // 0..127 == hidden unit
    #pragma unroll
    for (int r = 0; r < 2; ++r) {
        float acc = b1[t];
        for (int k = 0; k < 50; ++k)
            acc += features[r * 50 + k] * W1[k * HIDDEN + t];
        #pragma unroll
        for (int k = 0; k < 3; ++k)
            acc += latent[r * 3 + k] * W1[(50 + k) * HIDDEN + t];
        h_s[r * HIDDEN + t] = fmaxf(acc, 0.0f);      // relu
    }
    __syncthreads();
    if (t < 6) {                                     // scalars_per_domain (2x3)
        const int r = t / 3, c = t % 3;
        float acc = b2[c];
        for (int k = 0; k < HIDDEN; ++k)
            acc += h_s[r * HIDDEN + k] * W2[k * 3 + c];
        s_s[t] = acc;
    }
    __syncthreads();
    if (t < 6) {                                     // tv = s @ local_frame^T
        const int r = t / 3, c = t % 3;
        float acc = 0.0f;
        #pragma unroll
        for (int j = 0; j < 3; ++j)
            acc += s_s[r * 3 + j] * local_frame[c * 3 + j];
        tv_out[t] = acc;
    }
}

// ---------------------------------------------------------------------------
// Kernel B: per-residue window product + softmax -> attn; then
// trans = attn @ TV via V_WMMA_F32_16X16X4_F32 (two 16-residue tiles / wave).
// D-matrix is redistributed through LDS so every lane stores its residue's
// (attn0, attn1) and (tx, ty, tz) with contiguous, mergeable stores.
// ---------------------------------------------------------------------------
__global__ void __launch_bounds__(128)
attn_trans_kernel(const float* __restrict__ weights,
                  const float* __restrict__ tv,
                  float* __restrict__ out_attn,
                  float* __restrict__ out_trans,
                  int n_res) {
    const int lane = threadIdx.x & 31;
    const int wl   = threadIdx.x >> 5;               // wave index in block (0..3)
    const int wave = blockIdx.x * 4 + wl;
    const int base = wave * 32;
    const int i    = base + lane;
    const int ic   = (i < n_res) ? i : (n_res - 1);  // clamp: EXEC stays full

    // start = max(i-9,0); w0 = ceil(start/10)  (faithful index reconstruction)
    int s  = ic - 9; s = (s < 0) ? 0 : s;
    int w0 = s / 10 + ((s % 10) != 0 ? 1 : 0);

    // product over the 10-window gather (weights table is 1.6 MB -> L2 resident)
    float p0 = 1.0f, p1 = 1.0f;
    #pragma unroll
    for (int j = 0; j < 10; ++j) {
        const float* wp = weights + (size_t)(w0 + j) * 2;
        p0 *= wp[0];
        p1 *= wp[1];
    }

    // softmax over 2 domains
    const float m   = fmaxf(p0, p1);
    const float e0  = __expf(p0 - m);
    const float e1  = __expf(p1 - m);
    const float inv = 1.0f / (e0 + e1);
    const float a0  = e0 * inv;
    const float a1  = e1 * inv;

    // ---- B operand: TV (2x3) zero-padded into 4x16 -------------------------
    // vgpr0 = row K=0 (lanes 0-15) / K=2 (lanes 16-31); vgpr1 = K=1 / K=3.
    const float t00 = tv[0], t01 = tv[1], t02 = tv[2];
    const float t10 = tv[3], t11 = tv[4], t12 = tv[5];
    v2f B;
    B.x = (lane == 0) ? t00 : (lane == 1) ? t01 : (lane == 2) ? t02 : 0.0f;
    B.y = (lane == 0) ? t10 : (lane == 1) ? t11 : (lane == 2) ? t12 : 0.0f;

    // ---- A tile 1: residues base..base+15 ----------------------------------
    // vgpr0 = K=0 (lanes 0-15 hold M=0..15) / K=2; vgpr1 = K=1 / K=3.
    v2f A1;
    A1.x = (lane < 16) ? a0 : 0.0f;
    A1.y = (lane < 16) ? a1 : 0.0f;
    v8f C1 = {};
    C1 = __builtin_amdgcn_wmma_f32_16x16x4_f32(false, A1, false, B,
                                               (short)0, C1, false, false);

    // ---- A tile 2: residues base+16..base+31 (shuffle upper half down) -----
    const float a0h = __shfl(a0, (lane + 16) & 31, 32);
    const float a1h = __shfl(a1, (lane + 16) & 31, 32);
    v2f A2;
    A2.x = (lane < 16) ? a0h : 0.0f;
    A2.y = (lane < 16) ? a1h : 0.0f;
    v8f C2 = {};
    C2 = __builtin_amdgcn_wmma_f32_16x16x4_f32(false, A2, false, B,
                                               (short)0, C2, false, false);

    // ---- redistribute D through LDS: lane l <- residue base+l's 3 comps ----
    // C/D layout: vgpr r, lanes 0-15 -> M=r, N=lane; lanes 16-31 -> M=r+8.
    __shared__ float tr_s[4][32 * 3];
    const int comp = lane & 15;
    if (comp < 3) {                                  // lanes 0,1,2,16,17,18
        const int mOff = (lane >> 4) << 3;           // 0 or 8
        #pragma unroll
        for (int r = 0; r < 8; ++r) {
            tr_s[wl][(mOff + r) * 3 + comp]      = C1[r];
            tr_s[wl][(16 + mOff + r) * 3 + comp] = C2[r];
        }
    }
    __syncthreads();
    const float tx = tr_s[wl][lane * 3 + 0];
    const float ty = tr_s[wl][lane * 3 + 1];
    const float tz = tr_s[wl][lane * 3 + 2];

    // ---- coalesced per-lane stores -----------------------------------------
    if (i < n_res) {
        float* ap = out_attn + (size_t)i * 2;
        ap[0] = a0;
        ap[1] = a1;
        float* tp = out_trans + (size_t)i * 3;       // mergeable -> b96
        tp[0] = tx;
        tp[1] = ty;
        tp[2] = tz;
    }
}

// ---------------------------------------------------------------------------
// Kernel C: new_pos = pos + repeat(trans, 3).  float4 streaming add.
// 18M floats in + 18M out (+ L2-resident trans re-reads) -> pure HBM stream.
// ---------------------------------------------------------------------------
__global__ void apply_kernel(const float* __restrict__ pos,
                             const float* __restrict__ trans,
                             float* __restrict__ out_pos,
                             long n_flt) {
    const long k = (long)blockIdx.x * blockDim.x + threadIdx.x;  // float4 idx
    const long f = k * 4;
    if (f >= n_flt) return;
    const float4* p4 = (const float4*)pos;
    __builtin_prefetch(&p4[k + 4096], 0, 0);         // global_prefetch_b8
    if (f + 3 < n_flt) {
        float4 v = p4[k];
        float r[4] = {v.x, v.y, v.z, v.w};
        #pragma unroll
        for (int c = 0; c < 4; ++c) {
            const long ff  = f + c;
            const long res = ff / 9;                 // residue (const-div -> mul)
            const int  col = (int)(ff % 3);          // xyz component
            r[c] += trans[res * 3 + col];
        }
        float4 o; o.x = r[0]; o.y = r[1]; o.z = r[2]; o.w = r[3];
        ((float4*)out_pos)[k] = o;
    } else {                                         // scalar tail (unused here)
        for (long ff = f; ff < n_flt; ++ff)
            out_pos[ff] = pos[ff] + trans[(ff / 9) * 3 + (ff % 3)];
    }
}

// ---------------------------------------------------------------------------
extern "C" void kernel_launch(void* const* d_in, const int* in_sizes, int n_in,
                              void* d_out, int out_size, void* d_ws, size_t ws_size,
                              hipStream_t stream) {
    const float* weights  = (const float*)d_in[0];
    const float* latent   = (const float*)d_in[1];
    const float* features = (const float*)d_in[2];
    const float* W1       = (const float*)d_in[3];
    const float* b1       = (const float*)d_in[4];
    const float* W2       = (const float*)d_in[5];
    const float* b2       = (const float*)d_in[6];
    const float* lf       = (const float*)d_in[7];
    const float* pos      = (const float*)d_in[8];
    // d_in[9] = bs_per_res: indices recomputed in-kernel (saves 80 MB of reads)

    const long n_res = (long)in_sizes[8] / 9;        // positions are (3*N,3)

    float* out_pos   = (float*)d_out;                // 9*n_res
    float* out_attn  = out_pos + n_res * 9;          // 2*n_res
    float* out_trans = out_attn + n_res * 2;         // 3*n_res
    float* tv        = (float*)d_ws;                 // 6 floats scratch

    mlp_kernel<<<1, HIDDEN, 0, stream>>>(latent, features, W1, b1, W2, b2, lf, tv);

    // 4 waves (128 threads) per block; 32 residues per wave.
    const long waves   = (n_res + 31) / 32;
    const int  blocksB = (int)((waves + 3) / 4);
    attn_trans_kernel<<<blocksB, 128, 0, stream>>>(weights, tv, out_attn,
                                                   out_trans, (int)n_res);

    const long n_flt   = n_res * 9;
    const long quads   = (n_flt + 3) / 4;
    const int  blocksC = (int)((quads + 255) / 256);
    apply_kernel<<<blocksC, 256, 0, stream>>>(pos, out_trans, out_pos, n_flt);
}